// LIF_43559558316574
// MI455X (gfx1250) — compile-verified
//
#include <hip/hip_runtime.h>
#include <stdint.h>

// LIF forward (T=4, TAU=1.0, THRESH=1.0, hard reset) over x:[128,128,56,56] f32.
// Memory-bound streaming kernel: 411 MB total traffic, ~17.6 us floor @ 23.3 TB/s.
// Uses CDNA5 async global->LDS 128-bit loads (ASYNCcnt) for VGPR-free double
// buffering and non-temporal 128-bit stores.

typedef __attribute__((ext_vector_type(4))) float v4f;

#define THREADS 256
#define SLICE4  3211264u   // float4 elements per temporal slice (128*32*128*56*56/4 ... = 12,845,056/4)
#define TILES   12544u     // SLICE4 / THREADS
#define BLOCKS  3136u      // TILES / 4 -> each block handles 4 tiles

__global__ __launch_bounds__(THREADS) void LIF_43559558316574_kernel(
    const v4f* __restrict__ xv, v4f* __restrict__ ov)
{
    // 2 pipeline stages x 4 temporal slices x 256 lanes x 16B = 32 KB LDS.
    __shared__ v4f buf[2][4][THREADS];
    const unsigned tid = threadIdx.x;

    // Issue 4 async 128-bit global->LDS loads (one per temporal slice) for a tile.
    // Each lane loads into its own LDS slot and only ever reads its own slot back,
    // so no workgroup barrier is needed -- only s_wait_asynccnt (per-wave, in-order).
    auto issue = [&](unsigned tile, int stage) {
        const unsigned g = tile * THREADS + tid;
        #pragma unroll
        for (int s = 0; s < 4; ++s) {
            unsigned lds = (unsigned)(uintptr_t)(&buf[stage][s][tid]);
            uint64_t ga  = (uint64_t)(uintptr_t)(xv + (size_t)s * SLICE4 + g);
            asm volatile("global_load_async_to_lds_b128 %0, %1, off"
                         :: "v"(lds), "v"(ga)
                         : "memory");
        }
    };

    unsigned tile = blockIdx.x;
    const unsigned tstride = gridDim.x;
    int cur = 0;

    if (tile < TILES) issue(tile, cur);          // prologue prefetch

    for (; tile < TILES; tile += tstride) {
        const unsigned nxt = tile + tstride;
        if (nxt < TILES) {
            issue(nxt, cur ^ 1);
            // 8 async loads outstanding; loads complete in order, so <=4 means
            // the current tile's 4 loads have landed in LDS.
            asm volatile("s_wait_asynccnt 0x4" ::: "memory");
        } else {
            asm volatile("s_wait_asynccnt 0x0" ::: "memory");
        }

        // ds_load_b128 x4 from this lane's slots.
        v4f a0 = buf[cur][0][tid];
        v4f a1 = buf[cur][1][tid];
        v4f a2 = buf[cur][2][tid];
        v4f a3 = buf[cur][3][tid];

        v4f r0, r1, r2, r3;
        #pragma unroll
        for (int j = 0; j < 4; ++j) {
            // mem = mem*TAU + x_t ; spike = (mem >= 1) ; mem = (1-spike)*mem
            float m  = a0[j];
            float s0 = (m >= 1.0f) ? 1.0f : 0.0f;
            m = (1.0f - s0) * m;
            m += a1[j];
            float s1 = (m >= 1.0f) ? 1.0f : 0.0f;
            m = (1.0f - s1) * m;
            m += a2[j];
            float s2 = (m >= 1.0f) ? 1.0f : 0.0f;
            m = (1.0f - s2) * m;
            m += a3[j];
            float s3 = (m >= 1.0f) ? 1.0f : 0.0f;
            r0[j] = s0; r1[j] = s1; r2[j] = s2; r3[j] = s3;
        }

        // Write-once output: non-temporal 128-bit stores (don't pollute L2).
        const size_t g = (size_t)tile * THREADS + tid;
        __builtin_nontemporal_store(r0, ov + 0 * (size_t)SLICE4 + g);
        __builtin_nontemporal_store(r1, ov + 1 * (size_t)SLICE4 + g);
        __builtin_nontemporal_store(r2, ov + 2 * (size_t)SLICE4 + g);
        __builtin_nontemporal_store(r3, ov + 3 * (size_t)SLICE4 + g);

        cur ^= 1;
    }
}

extern "C" void kernel_launch(void* const* d_in, const int* in_sizes, int n_in,
                              void* d_out, int out_size, void* d_ws, size_t ws_size,
                              hipStream_t stream) {
    (void)in_sizes; (void)n_in; (void)out_size; (void)d_ws; (void)ws_size;
    const v4f* xv = (const v4f*)d_in[0];
    v4f*       ov = (v4f*)d_out;
    LIF_43559558316574_kernel<<<BLOCKS, THREADS, 0, stream>>>(xv, ov);
}